// Net_44255343018660
// MI455X (gfx1250) — compile-verified
//
#include <hip/hip_runtime.h>
#include <hip/hip_bf16.h>

typedef __attribute__((ext_vector_type(2))) float v2f;
typedef __attribute__((ext_vector_type(8))) float v8f;

// ---------------- degree / normalization ----------------

__global__ void k_deg_init(float* deg, int N) {
    int i = blockIdx.x * blockDim.x + threadIdx.x;
    if (i < N) deg[i] = 1.0f;              // self-loop contribution
}

__global__ void k_deg_edges(const int* __restrict__ col, float* deg, int E) {
    int e = blockIdx.x * blockDim.x + threadIdx.x;
    if (e < E) atomicAdd(&deg[col[e]], 1.0f);
}

// dis = rsqrt(deg) in place; s = self-loop term of layer-1 scalar aggregation
__global__ void k_dis_s(const float* __restrict__ x, float* dis, float* s, int N) {
    int i = blockIdx.x * blockDim.x + threadIdx.x;
    if (i < N) {
        float r = rsqrtf(dis[i]);
        dis[i] = r;
        s[i] = x[i] * r * r;
    }
}

// layer-1 aggregation is scalar because in_features == 1
__global__ void k_l1_edges(const int* __restrict__ row, const int* __restrict__ col,
                           const float* __restrict__ x, const float* __restrict__ dis,
                           float* s, int E) {
    int e = blockIdx.x * blockDim.x + threadIdx.x;
    if (e < E) {
        int r = row[e], c = col[e];
        atomicAdd(&s[c], x[r] * dis[r] * dis[c]);
    }
}

// ---------------- WMMA GEMM: hW2 = relu(s*W1 + b1) @ W2  [N,64] ----------------
// One wave32 per 16-node tile; A fragments synthesized on the fly (fused ReLU),
// B (W2, 256x64) staged in LDS. V_WMMA_F32_16X16X4_F32, K chunks of 4.

__global__ void __launch_bounds__(256)
k_gemm_wmma(const float* __restrict__ s, const float* __restrict__ W1,
            const float* __restrict__ b1, const float* __restrict__ W2,
            float* __restrict__ hW2, int N) {
    __shared__ float W2s[256 * 64];
    __shared__ float W1s[256];
    __shared__ float b1s[256];
    int tid = threadIdx.x;
    for (int i = tid; i < 256 * 64; i += 256) W2s[i] = W2[i];
    W1s[tid] = W1[tid];
    b1s[tid] = b1[tid];
    __syncthreads();

    int wave = tid >> 5;
    int lane = tid & 31;
    int base = (blockIdx.x * 8 + wave) * 16;     // 16 nodes per wave
    if (base < N) {                              // wave-uniform guard: EXEC stays all-ones
        int m     = lane & 15;                   // A row / B col / D col within tile
        int hi    = lane >> 4;                   // 0 or 1
        int koff2 = hi * 2;                      // K sub-offset {0,2}
        float sv  = s[base + m];

        #pragma unroll
        for (int nn = 0; nn < 4; ++nn) {         // 4 x 16 output columns
            v8f acc = {0.f, 0.f, 0.f, 0.f, 0.f, 0.f, 0.f, 0.f};
            for (int kk = 0; kk < 256; kk += 4) {
                int k0 = kk + koff2;
                // A[m][k0], A[m][k0+1] = relu(s*W1 + b1)
                float a0 = fmaxf(fmaf(sv, W1s[k0],     b1s[k0]),     0.0f);
                float a1 = fmaxf(fmaf(sv, W1s[k0 + 1], b1s[k0 + 1]), 0.0f);
                // B[k0][n], B[k0+1][n]
                float bb0 = W2s[k0 * 64 + nn * 16 + m];
                float bb1 = W2s[(k0 + 1) * 64 + nn * 16 + m];
                v2f A = {a0, a1};
                v2f B = {bb0, bb1};
                acc = __builtin_amdgcn_wmma_f32_16x16x4_f32(
                    /*neg_a=*/false, A, /*neg_b=*/false, B,
                    /*c_mod=*/(short)0, acc, /*reuse_a=*/false, /*reuse_b=*/false);
            }
            // D layout: VGPR r, lanes 0-15 -> M=r, lanes 16-31 -> M=r+8; N = lane&15
            int rbase = base + 8 * hi;
            int cidx  = nn * 16 + m;
            #pragma unroll
            for (int r = 0; r < 8; ++r)
                hW2[(size_t)(rbase + r) * 64 + cidx] = acc[r];
        }
    }
}

// ---------------- layer-2 aggregation ----------------

__global__ void k_agg2_init(const float* __restrict__ hW2, const float* __restrict__ dis,
                            float* agg2, long total) {
    long i = (long)blockIdx.x * blockDim.x + threadIdx.x;
    if (i < total) {
        int n = (int)(i >> 6);
        float r = dis[n];
        agg2[i] = hW2[i] * r * r;            // self-loop term
    }
}

// one wave per edge, 2 channels per lane: coalesced gather + f32 atomic scatter
__global__ void __launch_bounds__(256)
k_l2_edges(const int* __restrict__ row, const int* __restrict__ col,
           const float* __restrict__ dis, const float* __restrict__ hW2,
           float* agg2, int E) {
    int lane = threadIdx.x & 31;
    int e = blockIdx.x * 8 + (threadIdx.x >> 5);
    if (e < E) {
        int r = row[e], c = col[e];
        float nrm = dis[r] * dis[c];
        const float* src = hW2 + (size_t)r * 64;
        float* dst = agg2 + (size_t)c * 64;
        atomicAdd(&dst[lane],      src[lane]      * nrm);
        atomicAdd(&dst[lane + 32], src[lane + 32] * nrm);
    }
}

// ---------------- MLP head + log_softmax ----------------

__global__ void __launch_bounds__(256)
k_head(const float* __restrict__ agg2, const float* __restrict__ b2,
       const float* __restrict__ Wl1, const float* __restrict__ bl1,
       const float* __restrict__ Wl2, const float* __restrict__ bl2,
       float* __restrict__ out, int N) {
    __shared__ float sb2[64], sWl1[64 * 16], sbl1[16], sWl2[16 * 6], sbl2[6];
    int tid = threadIdx.x;
    for (int i = tid; i < 64;      i += 256) sb2[i]  = b2[i];
    for (int i = tid; i < 64 * 16; i += 256) sWl1[i] = Wl1[i];
    for (int i = tid; i < 16;      i += 256) sbl1[i] = bl1[i];
    for (int i = tid; i < 96;      i += 256) sWl2[i] = Wl2[i];
    for (int i = tid; i < 6;       i += 256) sbl2[i] = bl2[i];
    __syncthreads();

    int n = blockIdx.x * 256 + tid;
    if (n < N) {
        const float* a = agg2 + (size_t)n * 64;
        float h2[64];
        #pragma unroll
        for (int j = 0; j < 64; ++j) h2[j] = fmaxf(a[j] + sb2[j], 0.0f);

        float t[16];
        #pragma unroll
        for (int k = 0; k < 16; ++k) {
            float acc = sbl1[k];
            #pragma unroll
            for (int j = 0; j < 64; ++j) acc = fmaf(h2[j], sWl1[j * 16 + k], acc);
            t[k] = fmaxf(acc, 0.0f);
        }

        float o[6];
        float mx = -1e30f;
        #pragma unroll
        for (int c = 0; c < 6; ++c) {
            float acc = sbl2[c];
            #pragma unroll
            for (int k = 0; k < 16; ++k) acc = fmaf(t[k], sWl2[k * 6 + c], acc);
            o[c] = acc;
            mx = fmaxf(mx, acc);
        }
        float sum = 0.0f;
        #pragma unroll
        for (int c = 0; c < 6; ++c) sum += __expf(o[c] - mx);
        float lse = mx + __logf(sum);
        float* op = out + (size_t)n * 6;
        #pragma unroll
        for (int c = 0; c < 6; ++c) op[c] = o[c] - lse;
    }
}

// ---------------- launch ----------------

extern "C" void kernel_launch(void* const* d_in, const int* in_sizes, int n_in,
                              void* d_out, int out_size, void* d_ws, size_t ws_size,
                              hipStream_t stream) {
    const float* x   = (const float*)d_in[0];
    const int*   ei  = (const int*)  d_in[1];
    const float* W1  = (const float*)d_in[2];
    const float* b1  = (const float*)d_in[3];
    const float* W2  = (const float*)d_in[4];
    const float* b2  = (const float*)d_in[5];
    const float* Wl1 = (const float*)d_in[6];
    const float* bl1 = (const float*)d_in[7];
    const float* Wl2 = (const float*)d_in[8];
    const float* bl2 = (const float*)d_in[9];

    int N = in_sizes[0];          // x is [N,1]
    int E = in_sizes[1] / 2;      // edge_index is [2,E]
    const int* row  = ei;         // sources
    const int* colp = ei + E;     // targets (aggregation)

    float* ws   = (float*)d_ws;
    float* dis  = ws;                         // [N]   (deg, then rsqrt in place)
    float* sagg = ws + (size_t)N;             // [N]   layer-1 scalar aggregate
    float* hW2  = ws + 2 * (size_t)N;         // [N,64]
    float* agg2 = hW2 + (size_t)N * 64;       // [N,64]
    float* out  = (float*)d_out;              // [N,6]

    const int T = 256;
    k_deg_init <<<(N + T - 1) / T, T, 0, stream>>>(dis, N);
    k_deg_edges<<<(E + T - 1) / T, T, 0, stream>>>(colp, dis, E);
    k_dis_s    <<<(N + T - 1) / T, T, 0, stream>>>(x, dis, sagg, N);
    k_l1_edges <<<(E + T - 1) / T, T, 0, stream>>>(row, colp, x, dis, sagg, E);

    int tiles = (N + 15) / 16;
    k_gemm_wmma<<<(tiles + 7) / 8, 256, 0, stream>>>(sagg, W1, b1, W2, hW2, N);

    long total = (long)N * 64;
    k_agg2_init<<<(int)((total + T - 1) / T), T, 0, stream>>>(hW2, dis, agg2, total);
    k_l2_edges <<<(E + 7) / 8, 256, 0, stream>>>(row, colp, dis, hW2, agg2, E);

    k_head<<<(N + 255) / 256, 256, 0, stream>>>(agg2, b2, Wl1, bl1, Wl2, bl2, out, N);
}